// FCCaps_47184510714419
// MI455X (gfx1250) — compile-verified
//
#include <hip/hip_runtime.h>
#include <math.h>

// ---------------------------------------------------------------------------
// Problem constants (from the reference)
// ---------------------------------------------------------------------------
#define BATCH 256
#define IN_NC 2048      // input capsules
#define IN_DD 8         // input capsule dim
#define NCAP  16        // parent capsules
#define DCAP  16        // parent capsule dim
#define KDIM  (IN_NC * IN_DD)   // 16384 = GEMM reduction dim
#define COLS  (NCAP * DCAP)     // 256   = GEMM output cols (n*16+d)
#define KSEG  8                 // K split for parallelism
#define KCH   (KDIM / KSEG)     // 2048 per segment
#define KCHUNK 32               // K per LDS stage (8 WMMAs between barriers)

#define UBP 260                 // padded LDS row stride (floats): 260%16==4,
                                // 16B-aligned rows, bank stride 4 -> no conflicts

typedef float v2f  __attribute__((ext_vector_type(2)));
typedef float vf4  __attribute__((ext_vector_type(4)));
typedef float v8f  __attribute__((ext_vector_type(8)));

static __device__ __forceinline__ float dot4(vf4 a, vf4 b) {
  vf4 p = a * b;
  return p.x + p.y + p.z + p.w;
}

// ---------------------------------------------------------------------------
// Kernel 0: zero the workspace accumulators (ubar + s_acc), graph-safe.
// ---------------------------------------------------------------------------
__global__ void zero_ws_kernel(float* __restrict__ p, int n) {
  int idx = blockIdx.x * blockDim.x + threadIdx.x;
  if (idx < n) p[idx] = 0.0f;
}

// ---------------------------------------------------------------------------
// Kernel 1: ubar[b, n*16+d] = sum_K X[b,K] * W'[K, n*16+d]
//   GEMM M=256, N=256, K=16384 via v_wmma_f32_16x16x4_f32.
//   Grid: 16 b-tiles x 4 n-groups x 8 K-segments = 512 WGs of 4 waves.
//   Wave w owns one 16x16 tile (b-tile, n = ng*4+w).  K staged in 32-wide
//   LDS chunks (8 WMMAs per barrier pair); partials reduced with global
//   f32 atomics into zeroed ubar.  W is 16 MB -> lives in the 192 MB L2.
// ---------------------------------------------------------------------------
__global__ __launch_bounds__(128)
void ubar_gemm_wmma(const float* __restrict__ X,   // [256][16384]
                    const float* __restrict__ W,   // [16][16384][16] (flat)
                    float* __restrict__ ubar) {    // [256][256]
  __shared__ float At[16][36];        // 16 x 32 A tile, padded (36%16==4)
  __shared__ float Bt[4][KCHUNK * 16];// per-wave B tile (contiguous in mem)

  const int tid  = threadIdx.x;
  const int lane = tid & 31;
  const int w    = tid >> 5;

  const int bid  = blockIdx.x;
  const int kseg = bid & 7;
  const int ng   = (bid >> 3) & 3;
  const int bt   = bid >> 5;                 // 0..15 batch tile
  const int n    = ng * 4 + w;               // parent capsule = column block

  const float* Arow = X + (size_t)(bt * 16) * KDIM;
  const float* Wn   = W + (size_t)n * (KDIM * DCAP);   // contiguous [16384][16]

  const int M    = lane & 15;
  const int halo = lane >> 4;                // 0: K pair {0,1}; 1: K pair {2,3}
  const int koff = halo * 2;

  v8f acc = {};

  const int k_begin = kseg * KCH;
  const int k_end   = k_begin + KCH;
  for (int k0 = k_begin; k0 < k_end; k0 += KCHUNK) {
    // --- cooperative A tile load: 16 rows x 32 K (128B per row, coalesced) ---
    {
      int r  = tid >> 3;            // 0..15
      int c0 = (tid & 7) * 4;       // 0..28
      const vf4 av = *(const vf4*)(Arow + (size_t)r * KDIM + k0 + c0);
      *(vf4*)(&At[r][c0]) = av;     // row stride 144B -> 16B aligned
    }
    // --- per-wave B tile: W[n] rows k0..k0+31 are ONE contiguous 2KB run ---
    {
      const vf4* src = (const vf4*)(Wn + (size_t)k0 * DCAP);
      vf4* dst = (vf4*)(&Bt[w][0]);
#pragma unroll
      for (int q = 0; q < 4; ++q)
        dst[lane * 4 + q] = src[lane * 4 + q];
    }
    __syncthreads();

    // --- 8 x (16x16x4) f32 WMMA steps covering this K-chunk of 32 ---
#pragma unroll
    for (int s = 0; s < 8; ++s) {
      const int ks = s * 4 + koff;
      v2f a;  a.x  = At[M][ks];            a.y  = At[M][ks + 1];
      v2f bf; bf.x = Bt[w][ks * 16 + M];   bf.y = Bt[w][(ks + 1) * 16 + M];
      acc = __builtin_amdgcn_wmma_f32_16x16x4_f32(
          /*neg_a=*/false, a, /*neg_b=*/false, bf,
          /*c_mod=*/(short)0, acc, /*reuse_a=*/false, /*reuse_b=*/false);
    }
    __syncthreads();
  }

  // C/D layout: VGPR r -> row M=r (lanes 0-15) or M=r+8 (lanes 16-31), col=lane&15
#pragma unroll
  for (int r = 0; r < 8; ++r) {
    int row = bt * 16 + r + halo * 8;
    int col = n * DCAP + M;
    unsafeAtomicAdd(&ubar[row * COLS + col], acc[r]);
  }
}

// ---------------------------------------------------------------------------
// Kernel 2: fused attention.  Per (b,i): recompute u[n,:] (8x16 proj, W row
// served from L2 and broadcast across the 16 b-threads sharing i), score
// c=dot(u,ubar)/4, softmax over n, +bias, accumulate s into LDS, flush once
// with global atomics.  WG = 16 b x 16 i threads, 4 i-iterations.
// Capsule order staggered per i-thread (n = (nn+il)&15): concurrent lanes
// hit distinct LDS atomic addresses; padded rows (UBP) kill bank conflicts.
// ---------------------------------------------------------------------------
__global__ __launch_bounds__(256)
void attn_fused(const float* __restrict__ X,     // [256][2048][8]
                const float* __restrict__ W,     // [16][2048][8][16]
                const float* __restrict__ bias,  // [16][2048]
                const float* __restrict__ ubar,  // ws: [256][256]
                float* __restrict__ s_acc) {     // ws: [256][256]
  __shared__ float ub[16][UBP];    // ubar rows for this b-tile
  __shared__ float sl[16][UBP];    // s accumulator

  const int tid = threadIdx.x;
  const int bl  = tid & 15;        // local batch
  const int il  = tid >> 4;        // local i
  const int bid = blockIdx.x;
  const int ic  = bid & 31;        // i-chunk (64 i's each)
  const int bt  = bid >> 5;        // batch tile (16 b's each)
  const int b   = bt * 16 + bl;

#pragma unroll
  for (int j = 0; j < 16; ++j) {
    ub[j][tid] = ubar[(bt * 16 + j) * COLS + tid];
    sl[j][tid] = 0.0f;
  }
  __syncthreads();

  for (int ii = 0; ii < 4; ++ii) {
    const int i = ic * 64 + ii * 16 + il;

    // load x[b,i,0..7]
    const vf4* xp = (const vf4*)(X + ((size_t)b * IN_NC + i) * IN_DD);
    vf4 xa = xp[0], xb = xp[1];
    float xs[8] = {xa.x, xa.y, xa.z, xa.w, xb.x, xb.y, xb.z, xb.w};

    const float* Wi = W + (size_t)i * (IN_DD * DCAP);  // + n*262144 per capsule

    // ---- pass A: scores c = dot(u[n,:], ubar[b,n,:]) / sqrt(D) ----
    float c[NCAP];
#pragma unroll
    for (int nn = 0; nn < NCAP; ++nn) {
      const int n = (nn + il) & 15;                    // staggered capsule
      const float* wr = Wi + (size_t)n * (KDIM * DCAP);
      vf4 u0 = {}, u1 = {}, u2 = {}, u3 = {};
#pragma unroll
      for (int z = 0; z < IN_DD; ++z) {
        const vf4* wv = (const vf4*)(wr + z * DCAP);
        float xz = xs[z];
        u0 += xz * wv[0]; u1 += xz * wv[1]; u2 += xz * wv[2]; u3 += xz * wv[3];
      }
      const vf4* gv = (const vf4*)(&ub[bl][n * DCAP]);
      c[nn] = 0.25f * (dot4(u0, gv[0]) + dot4(u1, gv[1]) +
                       dot4(u2, gv[2]) + dot4(u3, gv[3]));
    }

    // ---- softmax over the 16 parent capsules (in-thread), + bias ----
    float m = c[0];
#pragma unroll
    for (int nn = 1; nn < NCAP; ++nn) m = fmaxf(m, c[nn]);
    float e[NCAP], sum = 0.0f;
#pragma unroll
    for (int nn = 0; nn < NCAP; ++nn) { e[nn] = __expf(c[nn] - m); sum += e[nn]; }
    float inv = 1.0f / sum;
#pragma unroll
    for (int nn = 0; nn < NCAP; ++nn) {
      const int n = (nn + il) & 15;
      e[nn] = e[nn] * inv + bias[n * IN_NC + i];
    }

    // ---- pass B: recompute u and accumulate s[b,n,d] += coeff*u ----
#pragma unroll
    for (int nn = 0; nn < NCAP; ++nn) {
      const int n = (nn + il) & 15;
      const float* wr = Wi + (size_t)n * (KDIM * DCAP);
      vf4 u0 = {}, u1 = {}, u2 = {}, u3 = {};
#pragma unroll
      for (int z = 0; z < IN_DD; ++z) {
        const vf4* wv = (const vf4*)(wr + z * DCAP);
        float xz = xs[z];
        u0 += xz * wv[0]; u1 += xz * wv[1]; u2 += xz * wv[2]; u3 += xz * wv[3];
      }
      float cf = e[nn];
      float* sp = &sl[bl][n * DCAP];
      unsafeAtomicAdd(&sp[0],  cf * u0.x); unsafeAtomicAdd(&sp[1],  cf * u0.y);
      unsafeAtomicAdd(&sp[2],  cf * u0.z); unsafeAtomicAdd(&sp[3],  cf * u0.w);
      unsafeAtomicAdd(&sp[4],  cf * u1.x); unsafeAtomicAdd(&sp[5],  cf * u1.y);
      unsafeAtomicAdd(&sp[6],  cf * u1.z); unsafeAtomicAdd(&sp[7],  cf * u1.w);
      unsafeAtomicAdd(&sp[8],  cf * u2.x); unsafeAtomicAdd(&sp[9],  cf * u2.y);
      unsafeAtomicAdd(&sp[10], cf * u2.z); unsafeAtomicAdd(&sp[11], cf * u2.w);
      unsafeAtomicAdd(&sp[12], cf * u3.x); unsafeAtomicAdd(&sp[13], cf * u3.y);
      unsafeAtomicAdd(&sp[14], cf * u3.z); unsafeAtomicAdd(&sp[15], cf * u3.w);
    }
  }

  __syncthreads();
#pragma unroll
  for (int j = 0; j < 16; ++j)
    unsafeAtomicAdd(&s_acc[(bt * 16 + j) * COLS + tid], sl[j][tid]);
}

// ---------------------------------------------------------------------------
// Kernel 3: squash.  out[b,n,:] = (|s|^2/(1+|s|^2)) * s/|s|.  4096 rows.
// ---------------------------------------------------------------------------
__global__ __launch_bounds__(256)
void squash_out(const float* __restrict__ s_acc, float* __restrict__ out) {
  int idx = blockIdx.x * 256 + threadIdx.x;     // (b*16+n), 0..4095
  const vf4* sp = (const vf4*)(s_acc + (size_t)idx * DCAP);
  vf4 s0 = sp[0], s1 = sp[1], s2 = sp[2], s3 = sp[3];
  float sq = dot4(s0, s0) + dot4(s1, s1) + dot4(s2, s2) + dot4(s3, s3);
  float scale = (sq / (1.0f + sq)) * rsqrtf(sq);
  vf4* op = (vf4*)(out + (size_t)idx * DCAP);
  op[0] = s0 * scale; op[1] = s1 * scale;
  op[2] = s2 * scale; op[3] = s3 * scale;
}

// ---------------------------------------------------------------------------
// Launch
// ---------------------------------------------------------------------------
extern "C" void kernel_launch(void* const* d_in, const int* in_sizes, int n_in,
                              void* d_out, int out_size, void* d_ws, size_t ws_size,
                              hipStream_t stream) {
  const float* X    = (const float*)d_in[0];   // [256][2048][8]
  const float* W    = (const float*)d_in[1];   // [16][2048][8][16]
  const float* bias = (const float*)d_in[2];   // [16][2048]

  float* ubar  = (float*)d_ws;                 // [256][256]
  float* s_acc = ubar + BATCH * COLS;          // [256][256]
  float* out   = (float*)d_out;                // [256][16][16]

  const int ws_floats = 2 * BATCH * COLS;      // 131072
  zero_ws_kernel<<<(ws_floats + 255) / 256, 256, 0, stream>>>((float*)d_ws, ws_floats);
  ubar_gemm_wmma<<<16 * 4 * KSEG, 128, 0, stream>>>(X, W, ubar);
  attn_fused<<<16 * 32, 256, 0, stream>>>(X, W, bias, ubar, s_acc);
  squash_out<<<16, 256, 0, stream>>>(s_acc, out);
}